// DecoderRNN_77068893160173
// MI455X (gfx1250) — compile-verified
//
#include <hip/hip_runtime.h>
#include <hip/hip_bf16.h>
#include <math.h>

// ---------------------------------------------------------------------------
// CDNA5 (gfx1250) fp32 WMMA decoder (show-attend-tell style GRU).
// V_WMMA_F32_16X16X4_F32: A 16x4, B 4x16, C/D 16x16 f32 (exact fp32 GEMM).
// Fragment layout (ISA 7.12.2, 32-bit A 16x4): lanes 0-15 hold K=0 (v0) and
// K=1 (v1); lanes 16-31 hold K=2 / K=3; M = lane&15.  B mirrors with
// N = lane&15.  Both A (MxK row-major) and W (NxK row-major, used as W^T)
// are K-major, so operand addressing is identical.
// C/D: VGPR r -> M = r + 8*(lane>>4), N = lane&15.
//
// Each wave computes a 16x32 tile: two independent accumulator chains share
// one A fragment -> 2x WMMA ILP on the XDL pipe and half the A-side cache
// traffic.  Block = 8 waves (4M x 2N) = 64x64 tile.
//
// Streaming kernels (attention/mean) use float4 (b128) loads to sustain the
// 23.3 TB/s HBM roof; tanh uses gfx1250's V_TANH_F32 when the builtin exists.
// ---------------------------------------------------------------------------

typedef float v2f __attribute__((ext_vector_type(2)));
typedef float v8f __attribute__((ext_vector_type(8)));

#if defined(__has_builtin)
#  if __has_builtin(__builtin_amdgcn_wmma_f32_16x16x4_f32)
#    define HAS_WMMA_F32X4 1
#  endif
#  if __has_builtin(__builtin_amdgcn_tanhf)
#    define HAS_HW_TANH 1
#  endif
#endif
#ifndef HAS_WMMA_F32X4
#  define HAS_WMMA_F32X4 0
#endif
#ifndef HAS_HW_TANH
#  define HAS_HW_TANH 0
#endif

constexpr int kB    = 64;
constexpr int kP    = 196;
constexpr int kE    = 2048;
constexpr int kH    = 512;
constexpr int kEMB  = 512;
constexpr int kV    = 10000;
constexpr int kTCAP = 50;
constexpr int kTMAX = 49;   // T_CAP - 1

__device__ __forceinline__ float dsigmoid(float x) {
  return 1.0f / (1.0f + __expf(-x));
}

__device__ __forceinline__ float fast_tanh(float x) {
#if HAS_HW_TANH
  return __builtin_amdgcn_tanhf(x);    // v_tanh_f32
#else
  return tanhf(x);
#endif
}

// Accumulate NT (1 or 2) 16x16 tiles: C[m0:+16, n0 + 16*tI :+16] of A * W^T.
template <int NT>
__device__ __forceinline__ void gemm_tile_accN(const float* __restrict__ A,
                                               const float* __restrict__ W,
                                               int K, int m0, int n0, int lane,
                                               v8f* acc) {
#if HAS_WMMA_F32X4
  const int row = lane & 15;          // M for A-frag, N for B-frag
  const int kb  = (lane >> 4) << 1;   // 0 or 2: K-split across lane halves
  const float* __restrict__ Ap  = A + (size_t)(m0 + row) * K + kb;
  const float* __restrict__ Wp0 = W + (size_t)(n0 + row) * K + kb;
  const float* __restrict__ Wp1 = W + (size_t)(n0 + 16 + row) * K + kb;
#pragma unroll 4
  for (int k = 0; k < K; k += 4) {
    v2f a;  a.x  = Ap[k];   a.y  = Ap[k + 1];
    v2f b0; b0.x = Wp0[k];  b0.y = Wp0[k + 1];
    acc[0] = __builtin_amdgcn_wmma_f32_16x16x4_f32(
        false, a, false, b0, (short)0, acc[0], false, false);
    if (NT > 1) {
      v2f b1; b1.x = Wp1[k]; b1.y = Wp1[k + 1];
      acc[1] = __builtin_amdgcn_wmma_f32_16x16x4_f32(
          false, a, false, b1, (short)0, acc[1], false, false);
    }
  }
#else
  const int hi = lane >> 4, col = lane & 15;
  for (int k = 0; k < K; ++k) {
#pragma unroll
    for (int tI = 0; tI < NT; ++tI) {
      float w = W[(size_t)(n0 + tI * 16 + col) * K + k];
#pragma unroll
      for (int r = 0; r < 8; ++r)
        acc[tI][r] += A[(size_t)(m0 + r + 8 * hi) * K + k] * w;
    }
  }
#endif
}

// Generic C[M,N] = A[M,K] * W[N,K]^T + bias[N].
// 256 threads = 8 waves (4M x 2N), wave tile 16x32, block tile 64x64.
__global__ __launch_bounds__(256) void gemm_wt_kernel(
    const float* __restrict__ A, const float* __restrict__ W,
    const float* __restrict__ bias, float* __restrict__ C,
    int M, int N, int K) {
  const int wave = threadIdx.x >> 5;
  const int lane = threadIdx.x & 31;
  const int m0 = blockIdx.x * 64 + (wave & 3) * 16;
  const int n0 = blockIdx.y * 64 + (wave >> 2) * 32;
  if (m0 >= M || n0 >= N) return;             // wave-uniform
  const int hi = lane >> 4, col = lane & 15;
  if (n0 + 16 < N) {                          // full 16x32 wave tile
    v8f acc[2] = {};
    gemm_tile_accN<2>(A, W, K, m0, n0, lane, acc);
#pragma unroll
    for (int tI = 0; tI < 2; ++tI) {
      const int n = n0 + tI * 16 + col;
      const float bv = bias[n];
#pragma unroll
      for (int r = 0; r < 8; ++r)
        C[(size_t)(m0 + hi * 8 + r) * N + n] = acc[tI][r] + bv;
    }
  } else {                                    // N-tail: single 16x16 tile
    v8f acc[1] = {};
    gemm_tile_accN<1>(A, W, K, m0, n0, lane, acc);
    const int n = n0 + col;
    const float bv = bias[n];
#pragma unroll
    for (int r = 0; r < 8; ++r)
      C[(size_t)(m0 + hi * 8 + r) * N + n] = acc[0][r] + bv;
  }
}

// preds[b, t, :] = mask(h_new @ fc1_w^T + fc1_b).  M = 64 fixed, N = 10000.
__global__ __launch_bounds__(256) void pred_gemm_kernel(
    const float* __restrict__ A,     // h_new (64 x 512)
    const float* __restrict__ W,     // fc1_w (10000 x 512)
    const float* __restrict__ bias,  // fc1_b
    float* __restrict__ preds,       // (B, TCAP, V)
    const int* __restrict__ lengths, int t) {
  const int wave = threadIdx.x >> 5;
  const int lane = threadIdx.x & 31;
  const int m0 = (wave & 3) * 16;
  const int n0 = blockIdx.y * 64 + (wave >> 2) * 32;
  if (n0 >= kV) return;
  const int hi = lane >> 4, col = lane & 15;
  if (n0 + 16 < kV) {
    v8f acc[2] = {};
    gemm_tile_accN<2>(A, W, kH, m0, n0, lane, acc);
#pragma unroll
    for (int tI = 0; tI < 2; ++tI) {
      const int n = n0 + tI * 16 + col;
      const float bv = bias[n];
#pragma unroll
      for (int r = 0; r < 8; ++r) {
        const int b = m0 + hi * 8 + r;
        const bool active = (lengths[b] - 1) >= t;
        preds[((size_t)b * kTCAP + t) * kV + n] =
            active ? (acc[tI][r] + bv) : 0.0f;
      }
    }
  } else {                                    // tail tile at n0 = 9984
    v8f acc[1] = {};
    gemm_tile_accN<1>(A, W, kH, m0, n0, lane, acc);
    const int n = n0 + col;
    const float bv = bias[n];
#pragma unroll
    for (int r = 0; r < 8; ++r) {
      const int b = m0 + hi * 8 + r;
      const bool active = (lengths[b] - 1) >= t;
      preds[((size_t)b * kTCAP + t) * kV + n] =
          active ? (acc[0][r] + bv) : 0.0f;
    }
  }
}

// fmean[b, e] = mean_p features[b, p, e]    (float4 lanes; grid (B, E/1024))
__global__ __launch_bounds__(256) void mean_feat_kernel(
    const float* __restrict__ features, float* __restrict__ fmean) {
  const int b = blockIdx.x;
  const int e4 = blockIdx.y * 256 + threadIdx.x;          // float4 index
  const float4* __restrict__ f =
      (const float4*)(features + (size_t)b * kP * kE) + e4;
  float4 s = make_float4(0.f, 0.f, 0.f, 0.f);
#pragma unroll 4
  for (int p = 0; p < kP; ++p) {
    const float4 v = f[(size_t)p * (kE / 4)];
    s.x += v.x; s.y += v.y; s.z += v.z; s.w += v.w;
  }
  const float inv = 1.0f / (float)kP;
  s.x *= inv; s.y *= inv; s.z *= inv; s.w *= inv;
  ((float4*)(fmean + (size_t)b * kE))[e4] = s;
}

// lam[b, p] = V_b + sum_e V_w[e] * tanh(x_wx[b,p,e] + h_wh[b,e])
// one wave per p (float4 lanes); 8 waves per block.
__global__ __launch_bounds__(256) void attn_lam_kernel(
    const float* __restrict__ xwx, const float* __restrict__ hwh,
    const float* __restrict__ Vw, const float* __restrict__ Vb,
    float* __restrict__ lam) {
  const int b = blockIdx.x;
  const int wave = threadIdx.x >> 5;
  const int lane = threadIdx.x & 31;
  const int p = blockIdx.y * 8 + wave;
  if (p >= kP) return;
  const float4* __restrict__ xr =
      (const float4*)(xwx + ((size_t)b * kP + p) * kE);
  const float4* __restrict__ hr = (const float4*)(hwh + (size_t)b * kE);
  const float4* __restrict__ vw = (const float4*)Vw;
  float s = 0.0f;
#pragma unroll 4
  for (int i = lane; i < kE / 4; i += 32) {               // 16 iterations
    const float4 x = xr[i];
    const float4 hh = hr[i];
    const float4 w = vw[i];
    s += w.x * fast_tanh(x.x + hh.x);
    s += w.y * fast_tanh(x.y + hh.y);
    s += w.z * fast_tanh(x.z + hh.z);
    s += w.w * fast_tanh(x.w + hh.w);
  }
#pragma unroll
  for (int off = 16; off > 0; off >>= 1) s += __shfl_xor(s, off, 32);
  if (lane == 0) lam[b * kP + p] = s + Vb[0];
}

// softmax over p (196) per b; alpha kept in ws, masked copy to alphas output.
__global__ __launch_bounds__(256) void attn_softmax_kernel(
    const float* __restrict__ lam, float* __restrict__ alpha,
    float* __restrict__ alphas_out, const int* __restrict__ lengths, int t) {
  const int b = blockIdx.x;
  const int tid = threadIdx.x;
  __shared__ float red[256];
  const float v = (tid < kP) ? lam[b * kP + tid] : -3.0e38f;
  red[tid] = v;
  __syncthreads();
  for (int s = 128; s > 0; s >>= 1) {
    if (tid < s) red[tid] = fmaxf(red[tid], red[tid + s]);
    __syncthreads();
  }
  const float mx = red[0];
  __syncthreads();
  const float ex = (tid < kP) ? __expf(v - mx) : 0.0f;
  red[tid] = ex;
  __syncthreads();
  for (int s = 128; s > 0; s >>= 1) {
    if (tid < s) red[tid] += red[tid + s];
    __syncthreads();
  }
  const float a = ex / red[0];
  if (tid < kP) {
    alpha[b * kP + tid] = a;
    const bool active = (lengths[b] - 1) >= t;
    alphas_out[((size_t)b * kTCAP + t) * kP + tid] = active ? a : 0.0f;
  }
}

// z[b, e] = sum_p alpha[b,p] * features[b,p,e]   (float4 lanes; grid (B, 2))
__global__ __launch_bounds__(256) void attn_z_kernel(
    const float* __restrict__ features, const float* __restrict__ alpha,
    float* __restrict__ z) {
  const int b = blockIdx.x;
  const int e4 = blockIdx.y * 256 + threadIdx.x;          // float4 index
  const float4* __restrict__ f =
      (const float4*)(features + (size_t)b * kP * kE) + e4;
  const float* __restrict__ al = alpha + b * kP;
  float4 s = make_float4(0.f, 0.f, 0.f, 0.f);
#pragma unroll 4
  for (int p = 0; p < kP; ++p) {
    const float a = al[p];
    const float4 v = f[(size_t)p * (kE / 4)];
    s.x += a * v.x; s.y += a * v.y; s.z += a * v.z; s.w += a * v.w;
  }
  ((float4*)(z + (size_t)b * kE))[e4] = s;
}

// inp[b, :2048] = sigmoid(gate_pre) * z ; inp[b, 2048:] = embed_w[captions[b,t]]
__global__ __launch_bounds__(256) void build_inp_kernel(
    const float* __restrict__ gate_pre, const float* __restrict__ z,
    const float* __restrict__ embed_w, const int* __restrict__ captions,
    int t, float* __restrict__ inp) {
  const int i = blockIdx.x * 256 + threadIdx.x;
  if (i >= kB * (kE + kEMB)) return;
  const int b = i / (kE + kEMB);
  const int j = i - b * (kE + kEMB);
  float v;
  if (j < kE) {
    v = dsigmoid(gate_pre[(size_t)b * kE + j]) * z[(size_t)b * kE + j];
  } else {
    const int tok = captions[b * kTCAP + t];
    v = embed_w[(size_t)tok * kEMB + (j - kE)];
  }
  inp[i] = v;
}

// GRU gate fusion + masked h update.
__global__ __launch_bounds__(256) void gru_update_kernel(
    const float* __restrict__ gi, const float* __restrict__ gh,
    float* __restrict__ h, float* __restrict__ h_new,
    const int* __restrict__ lengths, int t) {
  const int i = blockIdx.x * 256 + threadIdx.x;
  if (i >= kB * kH) return;
  const int b = i / kH;
  const int j = i - b * kH;
  const float* __restrict__ gib = gi + (size_t)b * 3 * kH;
  const float* __restrict__ ghb = gh + (size_t)b * 3 * kH;
  const float r  = dsigmoid(gib[j] + ghb[j]);
  const float zg = dsigmoid(gib[kH + j] + ghb[kH + j]);
  const float n  = fast_tanh(gib[2 * kH + j] + r * ghb[2 * kH + j]);
  const float hv = h[i];
  const float hn = (1.0f - zg) * n + zg * hv;
  h_new[i] = hn;
  if ((lengths[b] - 1) >= t) h[i] = hn;
}

// Zero preds+alphas region of d_out; write dec (= lengths - 1) as float.
__global__ __launch_bounds__(256) void init_out_kernel(
    float* __restrict__ out, const int* __restrict__ lengths) {
  const size_t N_PA = (size_t)kB * kTCAP * kV + (size_t)kB * kTCAP * kP;
  const size_t i = (size_t)blockIdx.x * 256 + threadIdx.x;
  if (i < N_PA) out[i] = 0.0f;
  if (i < kB) out[N_PA + i] = (float)(lengths[i] - 1);
}

// ---------------------------------------------------------------------------

extern "C" void kernel_launch(void* const* d_in, const int* in_sizes, int n_in,
                              void* d_out, int out_size, void* d_ws, size_t ws_size,
                              hipStream_t stream) {
  (void)in_sizes; (void)n_in; (void)out_size; (void)ws_size;

  const float* features = (const float*)d_in[0];
  const int*   captions = (const int*)d_in[1];
  const int*   lengths  = (const int*)d_in[2];
  const float* Wx_w     = (const float*)d_in[3];
  const float* Wx_b     = (const float*)d_in[4];
  const float* Wh_w     = (const float*)d_in[5];
  const float* Wh_b     = (const float*)d_in[6];
  const float* V_w      = (const float*)d_in[7];
  const float* V_b      = (const float*)d_in[8];
  const float* init_h_w = (const float*)d_in[9];
  const float* init_h_b = (const float*)d_in[10];
  const float* f_beta_w = (const float*)d_in[11];
  const float* f_beta_b = (const float*)d_in[12];
  const float* embed_w  = (const float*)d_in[13];
  const float* gru_w_ih = (const float*)d_in[14];
  const float* gru_b_ih = (const float*)d_in[15];
  const float* gru_w_hh = (const float*)d_in[16];
  const float* gru_b_hh = (const float*)d_in[17];
  const float* fc1_w    = (const float*)d_in[18];
  const float* fc1_b    = (const float*)d_in[19];

  float* out = (float*)d_out;
  float* preds_out  = out;                                   // B*TCAP*V
  float* alphas_out = out + (size_t)kB * kTCAP * kV;         // B*TCAP*P

  // Workspace layout (floats), ~106.6 MB total.
  float* ws = (float*)d_ws;
  size_t off = 0;
  float* xwx   = ws + off; off += (size_t)kB * kP * kE;      // 25,690,112
  float* fmean = ws + off; off += (size_t)kB * kE;
  float* h     = ws + off; off += (size_t)kB * kH;
  float* hnew  = ws + off; off += (size_t)kB * kH;
  float* hwh   = ws + off; off += (size_t)kB * kE;
  float* gatep = ws + off; off += (size_t)kB * kE;
  float* gh    = ws + off; off += (size_t)kB * 3 * kH;
  float* gi    = ws + off; off += (size_t)kB * 3 * kH;
  float* alpha = ws + off; off += (size_t)kB * kP;           // lam -> alpha in place
  float* z     = ws + off; off += (size_t)kB * kE;
  float* inp   = ws + off; off += (size_t)kB * (kE + kEMB);

  // ---- one-time setup ------------------------------------------------------
  {
    const size_t N_PA = (size_t)kB * kTCAP * kV + (size_t)kB * kTCAP * kP;
    const int nblk = (int)((N_PA + 255) / 256);
    init_out_kernel<<<nblk, 256, 0, stream>>>(out, lengths);
  }
  mean_feat_kernel<<<dim3(kB, kE / 1024), 256, 0, stream>>>(features, fmean);
  // h0 = fmean @ init_h_w^T + b    (64 x 512, K=2048)
  gemm_wt_kernel<<<dim3(1, kH / 64), 256, 0, stream>>>(
      fmean, init_h_w, init_h_b, h, kB, kH, kE);
  // x_wx = features @ Wx_w^T + b   (12544 x 2048, K=2048)  — dominant GEMM
  gemm_wt_kernel<<<dim3((kB * kP) / 64, kE / 64), 256, 0, stream>>>(
      features, Wx_w, Wx_b, xwx, kB * kP, kE, kE);

  // ---- recurrence ----------------------------------------------------------
  for (int t = 0; t < kTMAX; ++t) {
    // h-projections (all K=512, M=64)
    gemm_wt_kernel<<<dim3(1, kE / 64), 256, 0, stream>>>(
        h, Wh_w, Wh_b, hwh, kB, kE, kH);
    gemm_wt_kernel<<<dim3(1, kE / 64), 256, 0, stream>>>(
        h, f_beta_w, f_beta_b, gatep, kB, kE, kH);
    gemm_wt_kernel<<<dim3(1, (3 * kH) / 64), 256, 0, stream>>>(
        h, gru_w_hh, gru_b_hh, gh, kB, 3 * kH, kH);

    // attention
    attn_lam_kernel<<<dim3(kB, (kP + 7) / 8), 256, 0, stream>>>(
        xwx, hwh, V_w, V_b, alpha);
    attn_softmax_kernel<<<kB, 256, 0, stream>>>(
        alpha, alpha, alphas_out, lengths, t);
    attn_z_kernel<<<dim3(kB, kE / 1024), 256, 0, stream>>>(
        features, alpha, z);

    // GRU input + gates
    build_inp_kernel<<<(kB * (kE + kEMB) + 255) / 256, 256, 0, stream>>>(
        gatep, z, embed_w, captions, t, inp);
    gemm_wt_kernel<<<dim3(1, (3 * kH) / 64), 256, 0, stream>>>(
        inp, gru_w_ih, gru_b_ih, gi, kB, 3 * kH, kE + kEMB);
    gru_update_kernel<<<(kB * kH + 255) / 256, 256, 0, stream>>>(
        gi, gh, h, hnew, lengths, t);

    // vocabulary projection (64 x 10000, K=512), masked write into d_out
    pred_gemm_kernel<<<dim3(1, (kV + 63) / 64), 256, 0, stream>>>(
        hnew, fc1_w, fc1_b, preds_out, lengths, t);
  }
}